// HeterogeneousGraphTransformer_26757646254094
// MI455X (gfx1250) — compile-verified
//
#include <hip/hip_runtime.h>
#include <math.h>

// ---------------------------------------------------------------------------
// MI455X (gfx1250) HGT layer, fp32 (matches reference precision).
//  Phase 1: upd = x @ W1[t] + b ; Q = x @ W3[t] + b     (LDS-tiled WMMA GEMM)
//  Phase 2 (per edge type, KV/EK buffers reused):
//     Kb[d] = x @ W4[et, t_src, d] + b   (d = dst type), Vb[d] likewise (W2)
//     EK    = ef @ W5 + b
//     edge kernel: per-head dot, softmax over 8 heads, atomic segment-sum
//  Phase 3: per-type LayerNorm -> d_out
// Workspace: upd + Q + Kb(2N) + Vb(2N) + EK(E)  ~ 820 MB fp32.
// ---------------------------------------------------------------------------

typedef __attribute__((ext_vector_type(2))) float v2f;
typedef __attribute__((ext_vector_type(8))) float v8f;
typedef __attribute__((ext_vector_type(4))) int   v4i;

#define HDIM 512
#define WAVE 32

// GEMM tiling: block 256 thr (8 waves), tile 128(M) x 64(N), K-step 16,
// double-buffered LDS. Wave (wm,wn) in 4x2 grid owns 32x32 = 2x2 WMMA tiles.
#define BM 128
#define BN 64
#define BK 16
#define APAD 4   // A row stride 20 floats: keeps b128 stores 16B-aligned, spreads banks
#define BPAD 4   // B row stride 68 floats

// --- CDNA5 async global->LDS staging (ASYNCcnt-tracked), with fallback ------
#if __has_builtin(__builtin_amdgcn_global_load_async_to_lds_b128)
#define HAVE_ASYNC_LDS 1
typedef __attribute__((address_space(1))) v4i* gas_v4i_p;   // global (__device__)
typedef __attribute__((address_space(3))) v4i* las_v4i_p;   // LDS (__shared__)
#endif

__device__ __forceinline__ void stage16(const float* __restrict__ g, float* l)
{
#if defined(HAVE_ASYNC_LDS)
    // generic LDS pointers carry the LDS offset in the low 32 bits
    __builtin_amdgcn_global_load_async_to_lds_b128(
        (gas_v4i_p)(uintptr_t)g, (las_v4i_p)(uintptr_t)l, 0, 0);
#else
    *(float4*)l = *(const float4*)g;
#endif
}

__device__ __forceinline__ void async_fence()
{
#if defined(HAVE_ASYNC_LDS)
#if __has_builtin(__builtin_amdgcn_s_wait_asynccnt)
    __builtin_amdgcn_s_wait_asynccnt(0);
#else
    asm volatile("s_wait_asynccnt 0" ::: "memory");
#endif
#endif
}

__global__ __launch_bounds__(256)
void gemm_wmma_f32(const float* __restrict__ A, int lda,
                   const float* __restrict__ B, int ldb,
                   const float* __restrict__ bias,
                   float* __restrict__ C, int ldc,
                   int M, int Nc, int K)
{
    __shared__ float As[2][BM][BK + APAD];   // 2 x 128 x 20
    __shared__ float Bs[2][BK][BN + BPAD];   // 2 x 16 x 68

    const int t    = threadIdx.x;
    const int lane = t & (WAVE - 1);
    const int wid  = t >> 5;
    const int wm   = wid & 3;             // 0..3 -> 32-row group
    const int wn   = wid >> 2;            // 0..1 -> 32-col group
    const int half = lane >> 4;           // K-half of the wave (ISA A/B layout)
    const int l15  = lane & 15;

    const int gridN = (Nc + BN - 1) / BN;
    const int bn = blockIdx.x % gridN;
    const int bm = blockIdx.x / gridN;

    const int rowBlock = bm * BM;
    const int colBlock = bn * BN;
    const int rowBase  = rowBlock + wm * 32;
    const int colBase  = colBlock + wn * 32;

    // staging coordinates (all 256 threads)
    const int ar = t >> 1;                 // A: 2 threads per row, 8 floats each
    const int ak = (t & 1) * 8;
    const int grow = rowBlock + ar;
    const float* __restrict__ Ag = A + (size_t)(grow < M ? grow : 0) * lda;

    const int bkr = t >> 4;                // B: thread -> (k row, 4 cols)
    const int bc4 = (t & 15) * 4;
    const int gcol = colBlock + bc4;
    const float* __restrict__ Bg0 = B + (gcol < Nc ? gcol : 0);

    const bool blockInterior = (rowBlock + BM <= M) && (colBlock + BN <= Nc);

    auto stage = [&](int bb, int k0) {
        if (blockInterior && (k0 + BK <= K)) {
            // fast path: unguarded b128 async copies
            stage16(Ag + k0 + ak,     &As[bb][ar][ak]);
            stage16(Ag + k0 + ak + 4, &As[bb][ar][ak + 4]);
            stage16(Bg0 + (size_t)(k0 + bkr) * ldb, &Bs[bb][bkr][bc4]);
        } else {
            // ragged path: per-element guards, zero-fill
            #pragma unroll
            for (int i = 0; i < 8; ++i) {
                const int gk = k0 + ak + i;
                As[bb][ar][ak + i] = (grow < M && gk < K) ? Ag[gk] : 0.0f;
            }
            const int gk = k0 + bkr;
            const float* __restrict__ Bg = B + (size_t)(gk < K ? gk : 0) * ldb;
            #pragma unroll
            for (int i = 0; i < 4; ++i) {
                const int c = gcol + i;
                Bs[bb][bkr][bc4 + i] = (gk < K && c < Nc) ? Bg[c] : 0.0f;
            }
        }
    };

    v8f acc00 = {}, acc01 = {}, acc10 = {}, acc11 = {};

    auto compute = [&](int bb) {
        #pragma unroll
        for (int kk = 0; kk < BK; kk += 4) {
            const int kf = kk + half * 2;
            v2f a0 = *(const v2f*)&As[bb][wm * 32 + l15     ][kf];
            v2f a1 = *(const v2f*)&As[bb][wm * 32 + 16 + l15][kf];
            v2f b0, b1;
            b0.x = Bs[bb][kf    ][wn * 32 + l15];
            b0.y = Bs[bb][kf + 1][wn * 32 + l15];
            b1.x = Bs[bb][kf    ][wn * 32 + 16 + l15];
            b1.y = Bs[bb][kf + 1][wn * 32 + 16 + l15];
            acc00 = __builtin_amdgcn_wmma_f32_16x16x4_f32(false, a0, false, b0, (short)0, acc00, false, false);
            acc01 = __builtin_amdgcn_wmma_f32_16x16x4_f32(false, a0, false, b1, (short)0, acc01, false, false);
            acc10 = __builtin_amdgcn_wmma_f32_16x16x4_f32(false, a1, false, b0, (short)0, acc10, false, false);
            acc11 = __builtin_amdgcn_wmma_f32_16x16x4_f32(false, a1, false, b1, (short)0, acc11, false, false);
        }
    };

    // double-buffered pipeline: one barrier per K-block
    stage(0, 0);
    async_fence();
    __syncthreads();

    int buf = 0;
    for (int k0 = 0; k0 < K; k0 += BK) {
        if (k0 + BK < K) stage(buf ^ 1, k0 + BK);   // overlap next-tile staging
        compute(buf);
        async_fence();
        __syncthreads();
        buf ^= 1;
    }

    // ---- store 2x2 tiles (C/D layout: VGPR r -> M=r, +8 for upper half) ----
    auto store_tile = [&](const v8f& acc, int rowT, int colT) {
        const int col = colT + l15;
        if (col < Nc) {
            const float bv = bias ? bias[col] : 0.0f;
            #pragma unroll
            for (int r = 0; r < 8; ++r) {
                const int row = rowT + r + half * 8;
                if (row < M) C[(size_t)row * ldc + col] = acc[r] + bv;
            }
        }
    };
    store_tile(acc00, rowBase,      colBase);
    store_tile(acc01, rowBase,      colBase + 16);
    store_tile(acc10, rowBase + 16, colBase);
    store_tile(acc11, rowBase + 16, colBase + 16);
}

// ---------------- Edge attention: wave per edge ------------------------------
__global__ __launch_bounds__(256)
void edge_attention(const int* __restrict__ ei, const float* __restrict__ EK,
                    const float* __restrict__ Q, const float* __restrict__ Kb,
                    const float* __restrict__ Vb, const int* __restrict__ ntype,
                    float* __restrict__ upd, int E, int N)
{
    const int lane = threadIdx.x & (WAVE - 1);
    const int e = blockIdx.x * 8 + (threadIdx.x >> 5);
    if (e >= E) return;

    const int src = ei[e];
    const int dst = ei[E + e];
    const int td  = ntype[dst];

    const int c0 = lane * 16;
    const float* __restrict__ qp = Q  + (size_t)dst * HDIM + c0;
    const float* __restrict__ kp = Kb + ((size_t)td * N + src) * HDIM + c0;
    const float* __restrict__ vp = Vb + ((size_t)td * N + src) * HDIM + c0;
    const float* __restrict__ ep = EK + (size_t)e * HDIM + c0;

    float vv[16];
    float partial = 0.0f;
    #pragma unroll
    for (int i = 0; i < 4; ++i) {
        float4 q4 = ((const float4*)qp)[i];
        float4 k4 = ((const float4*)kp)[i];
        float4 v4 = ((const float4*)vp)[i];
        float4 e4 = ((const float4*)ep)[i];
        partial += q4.x * (k4.x + e4.x) + q4.y * (k4.y + e4.y)
                 + q4.z * (k4.z + e4.z) + q4.w * (k4.w + e4.w);
        vv[4*i+0] = v4.x + e4.x;
        vv[4*i+1] = v4.y + e4.y;
        vv[4*i+2] = v4.z + e4.z;
        vv[4*i+3] = v4.w + e4.w;
    }

    partial += __shfl_xor(partial, 1, WAVE);
    partial += __shfl_xor(partial, 2, WAVE);
    const float score = partial * 0.125f;          // 1/sqrt(64)

    float m = score;
    m = fmaxf(m, __shfl_xor(m, 4,  WAVE));
    m = fmaxf(m, __shfl_xor(m, 8,  WAVE));
    m = fmaxf(m, __shfl_xor(m, 16, WAVE));
    const float ex = __expf(score - m);
    float s = ex;
    s += __shfl_xor(s, 4,  WAVE);
    s += __shfl_xor(s, 8,  WAVE);
    s += __shfl_xor(s, 16, WAVE);
    const float w = ex / s;

    float* __restrict__ out = upd + (size_t)dst * HDIM + c0;
    #pragma unroll
    for (int i = 0; i < 16; ++i)
        unsafeAtomicAdd(out + i, w * vv[i]);       // global_atomic_add_f32
}

// ---------------- LayerNorm (per-type gamma/beta), wave per row --------------
__global__ __launch_bounds__(256)
void layernorm_out(const float* __restrict__ upd, const float* __restrict__ g,
                   const float* __restrict__ b, float* __restrict__ out,
                   int N, int NA)
{
    const int lane = threadIdx.x & (WAVE - 1);
    const int row = blockIdx.x * 8 + (threadIdx.x >> 5);
    if (row >= N) return;

    const int c0 = lane * 16;
    const float* __restrict__ xp = upd + (size_t)row * HDIM + c0;

    float x[16];
    float sum = 0.0f;
    #pragma unroll
    for (int i = 0; i < 4; ++i) {
        float4 tv = ((const float4*)xp)[i];
        x[4*i+0] = tv.x; x[4*i+1] = tv.y; x[4*i+2] = tv.z; x[4*i+3] = tv.w;
        sum += tv.x + tv.y + tv.z + tv.w;
    }
    #pragma unroll
    for (int off = 1; off < WAVE; off <<= 1) sum += __shfl_xor(sum, off, WAVE);
    const float mu = sum * (1.0f / 512.0f);

    float vs = 0.0f;
    #pragma unroll
    for (int i = 0; i < 16; ++i) { const float d = x[i] - mu; vs += d * d; }
    #pragma unroll
    for (int off = 1; off < WAVE; off <<= 1) vs += __shfl_xor(vs, off, WAVE);
    const float rstd = rsqrtf(vs * (1.0f / 512.0f) + 1e-5f);

    const int ty = (row >= NA) ? 1 : 0;
    const float* __restrict__ gp = g + (size_t)ty * HDIM + c0;
    const float* __restrict__ bp = b + (size_t)ty * HDIM + c0;
    float* __restrict__ op = out + (size_t)row * HDIM + c0;
    #pragma unroll
    for (int i = 0; i < 16; ++i)
        op[i] = (x[i] - mu) * rstd * gp[i] + bp[i];
}

// ---------------------------------------------------------------------------
extern "C" void kernel_launch(void* const* d_in, const int* in_sizes, int n_in,
                              void* d_out, int out_size, void* d_ws, size_t ws_size,
                              hipStream_t stream)
{
    const float* x_a   = (const float*)d_in[0];
    const float* x_b   = (const float*)d_in[1];
    const int*   ei_pos= (const int*)  d_in[2];
    const float* ef_pos= (const float*)d_in[3];
    const int*   ei_int= (const int*)  d_in[4];
    const float* ef_int= (const float*)d_in[5];
    const int*   ntype = (const int*)  d_in[6];
    const float* W1_w  = (const float*)d_in[7];
    const float* W1_b  = (const float*)d_in[8];
    const float* W2_w  = (const float*)d_in[9];
    const float* W2_b  = (const float*)d_in[10];
    const float* W4_w  = (const float*)d_in[11];
    const float* W4_b  = (const float*)d_in[12];
    const float* W3_w  = (const float*)d_in[13];
    const float* W3_b  = (const float*)d_in[14];
    const float* W5p_w = (const float*)d_in[15];
    const float* W5p_b = (const float*)d_in[16];
    const float* W5i_w = (const float*)d_in[17];
    const float* W5i_b = (const float*)d_in[18];
    const float* ln_g  = (const float*)d_in[19];
    const float* ln_b  = (const float*)d_in[20];

    const int NA   = in_sizes[0] / HDIM;
    const int NB   = in_sizes[1] / HDIM;
    const int N    = NA + NB;
    const int Epos = in_sizes[2] / 2;
    const int Eint = in_sizes[4] / 2;
    const int Dpos = in_sizes[3] / Epos;   // 60
    const int Dint = in_sizes[5] / Eint;   // 32

    float* ws = (float*)d_ws;
    size_t o = 0;
    float* upd = ws + o; o += (size_t)N * HDIM;
    float* Qb  = ws + o; o += (size_t)N * HDIM;
    float* Kb  = ws + o; o += (size_t)2 * N * HDIM;
    float* Vb  = ws + o; o += (size_t)2 * N * HDIM;
    float* EKb = ws + o;                    // max(Epos,Eint)*HDIM, reused per type

    auto gemm = [&](const float* A, int lda, const float* B, const float* bias,
                    float* C, int M, int Nc, int K) {
        const int gm = (M  + BM - 1) / BM;
        const int gn = (Nc + BN - 1) / BN;
        gemm_wmma_f32<<<gm * gn, 256, 0, stream>>>(A, lda, B, Nc, bias, C, Nc, M, Nc, K);
    };

    const size_t HH = (size_t)HDIM * HDIM;

    // Phase 1: upd = W1(x), Q = W3(x)   (per node type)
    gemm(x_a, HDIM, W1_w,      W1_b,        upd,                   NA, HDIM, HDIM);
    gemm(x_b, HDIM, W1_w + HH, W1_b + HDIM, upd + (size_t)NA*HDIM, NB, HDIM, HDIM);
    gemm(x_a, HDIM, W3_w,      W3_b,        Qb,                    NA, HDIM, HDIM);
    gemm(x_b, HDIM, W3_w + HH, W3_b + HDIM, Qb  + (size_t)NA*HDIM, NB, HDIM, HDIM);

    // Phase 2: per edge type
    for (int et = 0; et < 2; ++et) {
        const int*   ei  = et == 0 ? ei_pos : ei_int;
        const float* ef  = et == 0 ? ef_pos : ef_int;
        const float* W5w = et == 0 ? W5p_w  : W5i_w;
        const float* W5b = et == 0 ? W5p_b  : W5i_b;
        const int    E   = et == 0 ? Epos   : Eint;
        const int    D   = et == 0 ? Dpos   : Dint;

        for (int d = 0; d < 2; ++d) {
            const size_t i0 = (size_t)((et * 2 + 0) * 2 + d);  // src type 0
            const size_t i1 = (size_t)((et * 2 + 1) * 2 + d);  // src type 1
            float* Kd = Kb + (size_t)d * N * HDIM;
            float* Vd = Vb + (size_t)d * N * HDIM;
            gemm(x_a, HDIM, W4_w + i0*HH, W4_b + i0*HDIM, Kd,                   NA, HDIM, HDIM);
            gemm(x_b, HDIM, W4_w + i1*HH, W4_b + i1*HDIM, Kd + (size_t)NA*HDIM, NB, HDIM, HDIM);
            gemm(x_a, HDIM, W2_w + i0*HH, W2_b + i0*HDIM, Vd,                   NA, HDIM, HDIM);
            gemm(x_b, HDIM, W2_w + i1*HH, W2_b + i1*HDIM, Vd + (size_t)NA*HDIM, NB, HDIM, HDIM);
        }

        gemm(ef, D, W5w, W5b, EKb, E, HDIM, D);   // ek = ef @ W5 + b

        edge_attention<<<(E + 7) / 8, 256, 0, stream>>>(ei, EKb, Qb, Kb, Vb,
                                                        ntype, upd, E, N);
    }

    // Phase 3: per-type LayerNorm -> output
    layernorm_out<<<(N + 7) / 8, 256, 0, stream>>>(upd, ln_g, ln_b,
                                                   (float*)d_out, N, NA);
}